// GAT_65670049955846
// MI455X (gfx1250) — compile-verified
//
#include <hip/hip_runtime.h>

// GAT layer for MI455X (gfx1250): fused flash-style attention using
// V_WMMA_F32_16X16X4_F32 (fp32 tensor path, matching the fp32 reference).
//
// N=4096 nodes, C=64 channels, B=4 batch.
//   h = per-node projection (HBM-bound, VALU kernel)
//   y = softmax(mask(leakyrelu(h@a_src^T + e_dst))) @ h   (WMMA flash kernel)

#define NN   4096
#define CCH  64
#define BB   4
#define DIST_T 0.5f
#define LALPHA 0.01f
#define NEGINF (-1.0e15f)

typedef __attribute__((ext_vector_type(2))) float v2f;
typedef __attribute__((ext_vector_type(8))) float v8f;

// ---------------------------------------------------------------------------
// Kernel 1: h[b,n,:] = x[b,n,:] @ proj_w[n] + proj_b[n]
//           e_dst[b,n] = h[b,n,:] . a_w[n, C:2C]
// One wave per node; lane owns channels (lane, lane+32).
// ---------------------------------------------------------------------------
__global__ __launch_bounds__(256)
void gat_proj_kernel(const float* __restrict__ x,
                     const float* __restrict__ proj_w,
                     const float* __restrict__ proj_b,
                     const float* __restrict__ a_w,
                     float* __restrict__ h,
                     float* __restrict__ e_dst) {
    const int lane = threadIdx.x & 31;
    const int wave = threadIdx.x >> 5;
    const int n    = blockIdx.x * 8 + wave;      // 512 blocks * 8 waves = 4096
    const int c0   = lane;
    const int c1   = lane + 32;

    const float* __restrict__ W = proj_w + (size_t)n * CCH * CCH;

    float acc[BB][2];
#pragma unroll
    for (int b = 0; b < BB; ++b) { acc[b][0] = 0.f; acc[b][1] = 0.f; }

    for (int k = 0; k < CCH; ++k) {
        const float w0 = W[k * CCH + c0];
        const float w1 = W[k * CCH + c1];
#pragma unroll
        for (int b = 0; b < BB; ++b) {
            const float xv = x[((size_t)b * NN + n) * CCH + k];
            acc[b][0] = fmaf(xv, w0, acc[b][0]);
            acc[b][1] = fmaf(xv, w1, acc[b][1]);
        }
    }

    const float bias0 = proj_b[(size_t)n * CCH + c0];
    const float bias1 = proj_b[(size_t)n * CCH + c1];
    const float ad0   = a_w[(size_t)n * (2 * CCH) + CCH + c0];
    const float ad1   = a_w[(size_t)n * (2 * CCH) + CCH + c1];

#pragma unroll
    for (int b = 0; b < BB; ++b) {
        const float h0 = acc[b][0] + bias0;
        const float h1 = acc[b][1] + bias1;
        h[((size_t)b * NN + n) * CCH + c0] = h0;
        h[((size_t)b * NN + n) * CCH + c1] = h1;
        float pe = h0 * ad0 + h1 * ad1;
        pe += __shfl_xor(pe, 1,  32);
        pe += __shfl_xor(pe, 2,  32);
        pe += __shfl_xor(pe, 4,  32);
        pe += __shfl_xor(pe, 8,  32);
        pe += __shfl_xor(pe, 16, 32);
        if (lane == 0) e_dst[(size_t)b * NN + n] = pe;
    }
}

// ---------------------------------------------------------------------------
// Kernel 2: flash-style masked attention, one wave per (b, 16-row i-tile).
// S tile   = H_i(16x64) @ a_src_jtile^T(64x16)  -> 16x V_WMMA_F32_16X16X4_F32
// Y accum += P(16x16)   @ H_j(16x64)            -> 16x V_WMMA_F32_16X16X4_F32
//
// fp32 WMMA fragment layouts (ISA 7.12.2), hi = lane>>4, n = lane&15:
//   A 16x4 : lane row = n; .x = A[n][4kk+2hi], .y = A[n][4kk+2hi+1]
//   B 4x16 : lane col = n; .x = B[4kk+2hi][n], .y = B[4kk+2hi+1][n]
//   C 16x16: c[v] = C[v + 8*hi][n], v = 0..7
// ---------------------------------------------------------------------------
__global__ __launch_bounds__(128)
void gat_attn_kernel(const float* __restrict__ h,
                     const float* __restrict__ e_dst,
                     const float* __restrict__ a_w,
                     const float* __restrict__ dist_mat,
                     float* __restrict__ y_out) {
    __shared__ float lds_p[4][16 * 16];   // per-wave P transpose buffer

    const int lane = threadIdx.x & 31;
    const int wave = threadIdx.x >> 5;
    const int tile = blockIdx.x * 4 + wave;   // 0 .. 1023
    const int b    = tile >> 8;               // batch
    const int i0   = (tile & 255) * 16;       // row-tile origin
    const int n    = lane & 15;
    const int hi   = lane >> 4;

    const float* __restrict__ hb = h + (size_t)b * NN * CCH;

    // --- H_i as 16 A-fragments (whole 16x64 tile lives in 32 VGPRs) ---
    v2f hA[16];
    {
        const float* __restrict__ hrow = hb + (size_t)(i0 + n) * CCH + 2 * hi;
#pragma unroll
        for (int kk = 0; kk < 16; ++kk) {
            hA[kk].x = hrow[4 * kk];
            hA[kk].y = hrow[4 * kk + 1];
        }
    }

    v8f y0 = {}, y1 = {}, y2 = {}, y3 = {};
    float m_run[8], l_run[8];
#pragma unroll
    for (int v = 0; v < 8; ++v) { m_run[v] = -1.0e30f; l_run[v] = 0.f; }

    // per-lane dist_mat row pointers (row = i0 + v + 8*hi, col offset n)
    const float* __restrict__ drow[8];
#pragma unroll
    for (int v = 0; v < 8; ++v)
        drow[v] = dist_mat + (size_t)(i0 + v + 8 * hi) * NN + n;

    const float* __restrict__ edb = e_dst + (size_t)b * NN + n;

    for (int jt = 0; jt < NN / 16; ++jt) {
        const int j0 = jt * 16;

        // ---- S = H_i @ a_src_j^T : B[k][n] = a_w[(j0+n)*2C + k] ----
        v8f s = {};
        const float* __restrict__ bs =
            a_w + (size_t)(j0 + n) * (2 * CCH) + 2 * hi;
#pragma unroll
        for (int kk = 0; kk < 16; ++kk) {
            v2f bf;
            bf.x = bs[4 * kk];
            bf.y = bs[4 * kk + 1];
            s = __builtin_amdgcn_wmma_f32_16x16x4_f32(
                    false, hA[kk], false, bf, (short)0, s, false, false);
        }

        const float ed = edb[j0];

        // ---- elementwise: +e_dst, leakyrelu, adjacency mask ----
        float p[8], rmax[8];
#pragma unroll
        for (int v = 0; v < 8; ++v) {
            float sv = s[v] + ed;
            sv = (sv >= 0.f) ? sv : LALPHA * sv;
            const float d  = drow[v][j0];
            const int   ig = i0 + v + 8 * hi;
            const int   jg = j0 + n;
            const bool adj = ((d > 0.f) && (d < DIST_T)) || (ig == jg);
            sv = adj ? sv : NEGINF;
            p[v] = sv;
            float rm = sv;                      // row max over 16 cols
            rm = fmaxf(rm, __shfl_xor(rm, 1, 32));
            rm = fmaxf(rm, __shfl_xor(rm, 2, 32));
            rm = fmaxf(rm, __shfl_xor(rm, 4, 32));
            rm = fmaxf(rm, __shfl_xor(rm, 8, 32));
            rmax[v] = rm;
        }

        // ---- online softmax update ----
        float alpha_s[8];
#pragma unroll
        for (int v = 0; v < 8; ++v) {
            const float mnew = fmaxf(m_run[v], rmax[v]);
            const float al   = __expf(m_run[v] - mnew);
            m_run[v] = mnew;
            float pv = __expf(p[v] - mnew);
            p[v] = pv;
            float rs = pv;                      // row sum over 16 cols
            rs += __shfl_xor(rs, 1, 32);
            rs += __shfl_xor(rs, 2, 32);
            rs += __shfl_xor(rs, 4, 32);
            rs += __shfl_xor(rs, 8, 32);
            l_run[v] = l_run[v] * al + rs;
            alpha_s[v] = al;
        }
#pragma unroll
        for (int v = 0; v < 8; ++v) {
            y0[v] *= alpha_s[v];
            y1[v] *= alpha_s[v];
            y2[v] *= alpha_s[v];
            y3[v] *= alpha_s[v];
        }

        // ---- transpose P (C-layout -> A-layout) through LDS ----
#pragma unroll
        for (int v = 0; v < 8; ++v)
            lds_p[wave][(v + 8 * hi) * 16 + n] = p[v];
        __builtin_amdgcn_wave_barrier();        // same-wave DS ops are in-order
        v2f ap[4];
#pragma unroll
        for (int kk = 0; kk < 4; ++kk) {
            ap[kk].x = lds_p[wave][n * 16 + 4 * kk + 2 * hi];
            ap[kk].y = lds_p[wave][n * 16 + 4 * kk + 2 * hi + 1];
        }
        __builtin_amdgcn_wave_barrier();

        // ---- Y += P @ H_j  (4 column tiles of 16) ----
        const float* __restrict__ hj = hb + (size_t)j0 * CCH + n;
#pragma unroll
        for (int kk = 0; kk < 4; ++kk) {
            const float* __restrict__ hr = hj + (size_t)(4 * kk + 2 * hi) * CCH;
            v2f b0, b1, b2, b3;
            b0.x = hr[0];  b0.y = hr[CCH + 0];
            b1.x = hr[16]; b1.y = hr[CCH + 16];
            b2.x = hr[32]; b2.y = hr[CCH + 32];
            b3.x = hr[48]; b3.y = hr[CCH + 48];
            y0 = __builtin_amdgcn_wmma_f32_16x16x4_f32(
                     false, ap[kk], false, b0, (short)0, y0, false, false);
            y1 = __builtin_amdgcn_wmma_f32_16x16x4_f32(
                     false, ap[kk], false, b1, (short)0, y1, false, false);
            y2 = __builtin_amdgcn_wmma_f32_16x16x4_f32(
                     false, ap[kk], false, b2, (short)0, y2, false, false);
            y3 = __builtin_amdgcn_wmma_f32_16x16x4_f32(
                     false, ap[kk], false, b3, (short)0, y3, false, false);
        }
    }

    // ---- finalize: divide by softmax denominator, store ----
#pragma unroll
    for (int v = 0; v < 8; ++v) {
        const float inv = 1.0f / l_run[v];
        float* __restrict__ orow =
            y_out + ((size_t)b * NN + i0 + v + 8 * hi) * CCH + n;
        orow[0]  = y0[v] * inv;
        orow[16] = y1[v] * inv;
        orow[32] = y2[v] * inv;
        orow[48] = y3[v] * inv;
    }
}

// ---------------------------------------------------------------------------
extern "C" void kernel_launch(void* const* d_in, const int* in_sizes, int n_in,
                              void* d_out, int out_size, void* d_ws, size_t ws_size,
                              hipStream_t stream) {
    const float* x    = (const float*)d_in[0];   // (B,N,C)
    const float* dist = (const float*)d_in[1];   // (N,N)
    const float* pw   = (const float*)d_in[2];   // (N,C,C)
    const float* pb   = (const float*)d_in[3];   // (N,C)
    const float* aw   = (const float*)d_in[4];   // (N,2C)

    float* h  = (float*)d_ws;                       // B*N*C floats (4 MB)
    float* ed = h + (size_t)BB * NN * CCH;          // B*N floats

    gat_proj_kernel<<<NN / 8, 256, 0, stream>>>(x, pw, pb, aw, h, ed);

    const int tiles = BB * (NN / 16);               // 1024 wave-tiles
    gat_attn_kernel<<<tiles / 4, 128, 0, stream>>>(h, ed, aw, dist,
                                                   (float*)d_out);
}